// BiLSTM_CRF_45260365365671
// MI455X (gfx1250) — compile-verified
//
#include <hip/hip_runtime.h>
#include <hip/hip_bf16.h>

typedef __attribute__((ext_vector_type(2))) float v2f;
typedef __attribute__((ext_vector_type(8))) float v8f;

#define TAGS 32
#define TAG_START 30
#define TAG_STOP 31

// ---------------------------------------------------------------------------
// Forward (partition function) kernel.
// One wave32 per 16 batch rows. Per time step:
//   m_b   = rowmax(alpha)
//   P     = exp(alpha - m)            (staged via double-buffered LDS tile)
//   D     = P (16x32) * E (32x32)     E = exp(transitions); K=32 split into
//                                     two independent 4-deep chains of
//                                     v_wmma_f32_16x16x4_f32 per 16x16 tile
//   alpha = m + log(D) + feats[:,t,:]
// last_alpha captured branchlessly at t == len-1; final logsumexp + atomicAdd.
// ---------------------------------------------------------------------------
__global__ __launch_bounds__(32) void crf_forward_kernel(
    const float* __restrict__ feats,
    const float* __restrict__ trans,
    const int*   __restrict__ lens,
    float*       __restrict__ out,
    int B, int L)
{
    const int lane = threadIdx.x;      // 0..31
    const int hi   = lane >> 4;        // half-wave select
    const int lc   = lane & 15;        // lane within half
    const int b0   = blockIdx.x * 16;  // 16 batch rows per wave

    __shared__ float E[TAGS][TAGS];    // exp(transitions)
    __shared__ float P[2][16][36];     // exp(alpha - m), double-buffered

    // Build E = exp(trans) cooperatively (single wave).
    for (int k = lane; k < TAGS * TAGS; k += 32)
        E[k >> 5][k & 31] = __expf(trans[k]);
    __syncthreads();

    // Persistent B fragments of E for v_wmma_f32_16x16x4_f32.
    // B (4x16, K x N): VGPR0/1 hold K = {0,1} for lanes 0-15, {2,3} for 16-31.
    v2f bfrag[8][2];
#pragma unroll
    for (int kk = 0; kk < 8; ++kk)
#pragma unroll
        for (int n = 0; n < 2; ++n) {
            const int k0  = 4 * kk + 2 * hi;
            const int col = 16 * n + lc;
            bfrag[kk][n][0] = E[k0][col];
            bfrag[kk][n][1] = E[k0 + 1][col];
        }

    // trans[:, STOP] in the C/D lane layout (col = 16n + lc).
    float tstop[2];
    tstop[0] = trans[(16 * 0 + lc) * TAGS + TAG_STOP];
    tstop[1] = trans[(16 * 1 + lc) * TAGS + TAG_STOP];

    // Per-VGPR-slot row bases and sequence lengths.
    // C/D layout: VGPR v <-> batch row (v + 8*hi) of the tile.
    size_t rb[8];
    int    lenm[8];
#pragma unroll
    for (int v = 0; v < 8; ++v) {
        const int row = v + 8 * hi;
        const int b   = b0 + row;
        rb[v]   = (size_t)b * L * TAGS + lc;  // add t*TAGS + 16n per access
        lenm[v] = lens[b];
    }

    // alpha0 = trans[START,:] + feats[:,0,:]
    float AL[2][8];   // current alpha, D layout
    float LA[2][8];   // captured last_alpha
#pragma unroll
    for (int n = 0; n < 2; ++n)
#pragma unroll
        for (int v = 0; v < 8; ++v) {
            const int col = 16 * n + lc;
            const float a0 = trans[TAG_START * TAGS + col] + feats[rb[v] + 16 * n];
            AL[n][v] = a0;
            LA[n][v] = a0;  // correct if len==1; overwritten at t==len-1 otherwise
        }

    for (int t = 1; t < L; ++t) {
        const int pb = t & 1;  // LDS double-buffer select

        // Prefetch feats rows ~16 steps ahead (global_prefetch_b8).
        if (t + 16 < L) {
#pragma unroll
            for (int v = 0; v < 8; ++v)
                __builtin_prefetch(&feats[rb[v] + (size_t)(t + 16) * TAGS], 0, 1);
        }

        // Issue this step's feats loads up front: independent of alpha, so
        // they overlap the max/exp/LDS/WMMA chain below.
        float fv[2][8];
#pragma unroll
        for (int n = 0; n < 2; ++n)
#pragma unroll
            for (int v = 0; v < 8; ++v)
                fv[n][v] = feats[rb[v] + (size_t)t * TAGS + 16 * n];

        // Row max over all 32 tags (both N-tiles, then across 16 lanes).
        float m[8];
#pragma unroll
        for (int v = 0; v < 8; ++v) {
            float mm = fmaxf(AL[0][v], AL[1][v]);
            mm = fmaxf(mm, __shfl_xor(mm, 8, 16));
            mm = fmaxf(mm, __shfl_xor(mm, 4, 16));
            mm = fmaxf(mm, __shfl_xor(mm, 2, 16));
            mm = fmaxf(mm, __shfl_xor(mm, 1, 16));
            m[v] = mm;
        }

        // P = exp(alpha - m) -> LDS buffer pb (D layout to row-major tile).
#pragma unroll
        for (int n = 0; n < 2; ++n)
#pragma unroll
            for (int v = 0; v < 8; ++v)
                P[pb][v + 8 * hi][16 * n + lc] = __expf(AL[n][v] - m[v]);
        __syncthreads();  // single-wave WG: lowers to a dscnt wait ordering reads

        // D = P * E : per N-tile, K=32 as two independent 4-deep WMMA chains.
        v8f acc[2][2];    // [n][half-chain]
#pragma unroll
        for (int n = 0; n < 2; ++n)
#pragma unroll
            for (int h = 0; h < 2; ++h)
#pragma unroll
                for (int i = 0; i < 8; ++i) acc[n][h][i] = 0.0f;

#pragma unroll
        for (int h = 0; h < 2; ++h) {
#pragma unroll
            for (int q = 0; q < 4; ++q) {
                const int kk = 4 * h + q;
                // A (16x4 f32): lane<16 -> M=lane, K={4kk,4kk+1}; lane>=16 -> +2
                const int k0 = 4 * kk + 2 * hi;
                v2f a;
                a[0] = P[pb][lc][k0];
                a[1] = P[pb][lc][k0 + 1];
                acc[0][h] = __builtin_amdgcn_wmma_f32_16x16x4_f32(
                    false, a, false, bfrag[kk][0], (short)0, acc[0][h], false, false);
                acc[1][h] = __builtin_amdgcn_wmma_f32_16x16x4_f32(
                    false, a, false, bfrag[kk][1], (short)0, acc[1][h], false, false);
            }
        }
        // No barrier here: double-buffered P removes the WAR hazard on LDS.

        // alpha = m + log(D) + feats[:,t,:]; capture at t == len-1.
#pragma unroll
        for (int n = 0; n < 2; ++n)
#pragma unroll
            for (int v = 0; v < 8; ++v) {
                const float d  = acc[n][0][v] + acc[n][1][v];
                const float na = m[v] + __logf(d) + fv[n][v];
                AL[n][v] = na;
                LA[n][v] = (t == lenm[v] - 1) ? na : LA[n][v];
            }
    }

    // forward_score contribution: sum_b logsumexp(last_alpha + trans[:,STOP]).
    float accum = 0.0f;
#pragma unroll
    for (int v = 0; v < 8; ++v) {
        const float la0 = LA[0][v] + tstop[0];
        const float la1 = LA[1][v] + tstop[1];
        float mx = fmaxf(la0, la1);
        mx = fmaxf(mx, __shfl_xor(mx, 8, 16));
        mx = fmaxf(mx, __shfl_xor(mx, 4, 16));
        mx = fmaxf(mx, __shfl_xor(mx, 2, 16));
        mx = fmaxf(mx, __shfl_xor(mx, 1, 16));
        float s = __expf(la0 - mx) + __expf(la1 - mx);
        s += __shfl_xor(s, 8, 16);
        s += __shfl_xor(s, 4, 16);
        s += __shfl_xor(s, 2, 16);
        s += __shfl_xor(s, 1, 16);
        accum += mx + __logf(s);
    }
    if (lc == 0)  // lanes 0 and 16 each own 8 distinct rows
        atomicAdd(out, accum);
}

// ---------------------------------------------------------------------------
// Gold-path score: one thread per sequence; feats are L2-resident after the
// forward kernel (64 MB << 192 MB L2). Subtracts gold score from the output.
// ---------------------------------------------------------------------------
__global__ __launch_bounds__(256) void crf_gold_kernel(
    const float* __restrict__ feats,
    const float* __restrict__ trans,
    const int*   __restrict__ tags,
    const int*   __restrict__ lens,
    float*       __restrict__ out,
    int B, int L)
{
    const int b = blockIdx.x * blockDim.x + threadIdx.x;
    if (b >= B) return;

    const int*   tg = tags  + (size_t)b * L;
    const float* fb = feats + (size_t)b * L * TAGS;

    const int t0  = tg[0];
    float g = trans[TAG_START * TAGS + t0] + fb[t0];           // begin
    const int len = lens[b];
    g += trans[tg[len - 1] * TAGS + TAG_STOP];                 // end

    int prev = t0;
    for (int t = 1; t < L; ++t) {
        const int cur = tg[t];
        if (cur != 0)                                          // mask: tag != 0
            g += trans[prev * TAGS + cur] + fb[(size_t)t * TAGS + cur];
        prev = cur;
    }
    atomicAdd(out, -g);
}

extern "C" void kernel_launch(void* const* d_in, const int* in_sizes, int n_in,
                              void* d_out, int out_size, void* d_ws, size_t ws_size,
                              hipStream_t stream) {
    const float* feats = (const float*)d_in[0];
    const float* trans = (const float*)d_in[1];
    const int*   tags  = (const int*)d_in[2];
    const int*   lens  = (const int*)d_in[3];
    float*       out   = (float*)d_out;

    const int B = in_sizes[3];
    const int L = in_sizes[2] / B;

    hipMemsetAsync(out, 0, sizeof(float), stream);
    crf_forward_kernel<<<dim3(B / 16), dim3(32), 0, stream>>>(feats, trans, lens, out, B, L);
    crf_gold_kernel<<<dim3((B + 255) / 256), dim3(256), 0, stream>>>(feats, trans, tags, lens, out, B, L);
}